// GNN_embedder_35124242547153
// MI455X (gfx1250) — compile-verified
//
#include <hip/hip_runtime.h>
#include <hip/hip_bf16.h>

typedef __attribute__((ext_vector_type(2))) float v2f;
typedef __attribute__((ext_vector_type(8))) float v8f;

#define NEG_SLOPE 0.2f

// ---------- helpers: monotone float<->uint key for atomic max over floats ----
__device__ __forceinline__ unsigned fkey_enc(float f) {
    unsigned b = __float_as_uint(f);
    return (b & 0x80000000u) ? ~b : (b | 0x80000000u);
}
__device__ __forceinline__ float fkey_dec(unsigned k) {
    return (k & 0x80000000u) ? __uint_as_float(k ^ 0x80000000u)
                             : __uint_as_float(~k);
}

// ---------- K0: zero scratch ------------------------------------------------
__global__ void k_zero(float* __restrict__ p, long long cnt) {
    long long i = (long long)blockIdx.x * blockDim.x + threadIdx.x;
    long long stride = (long long)gridDim.x * blockDim.x;
    for (; i < cnt; i += stride) p[i] = 0.0f;
}

// ---------- K1: fused NNConv edge GEMM via WMMA f32 16x16x4 -----------------
// msg[e,o] = sum_{f,i} ea[e,f]*x[src[e],i]*W_nn[f,i*16+o] + sum_i x_i*b_nn[i*16+o]
// => [16 edges,272] @ [272,16] per wave, K split into 68 chunks of 4.
__global__ void k_nnconv(const float* __restrict__ x,
                         const int*   __restrict__ eidx,
                         const float* __restrict__ eattr,
                         const float* __restrict__ W_nn,
                         const float* __restrict__ b_nn,
                         float* __restrict__ agg,
                         int N, int E) {
    __shared__ float sWp[136 * 32];      // W' pair-interleaved: [p=k>>1][o][k&1]
    __shared__ float sEA[8][16][16];     // per-wave edge_attr rows
    __shared__ float sXS[8][16][16];     // per-wave x[src] rows
    __shared__ int   sDST[8][16];

    const int tid = threadIdx.x;
    // Stage W' (272 x 16): k<256 -> W_nn[k>>4, (k&15)*16+o]; k>=256 -> b_nn
    for (int t = tid; t < 272 * 16; t += 256) {
        int k = t >> 4, o = t & 15;
        float v = (k < 256) ? W_nn[(k >> 4) * 256 + (k & 15) * 16 + o]
                            : b_nn[(k - 256) * 16 + o];
        sWp[(k >> 1) * 32 + o * 2 + (k & 1)] = v;
    }

    const int wave = tid >> 5;
    const int lane = tid & 31;
    const int m    = lane & 15;   // row within tile (A) / col (B,C)
    const int hi   = lane >> 4;   // K-half selector
    const int base = (blockIdx.x * 8 + wave) * 16;
    const int e    = base + m;
    const bool valid = (e < E);
    const int ec = valid ? e : 0;

    if (hi == 0) {
        for (int j = 0; j < 16; ++j)
            sEA[wave][m][j] = valid ? eattr[(size_t)ec * 16 + j] : 0.0f;
        sDST[wave][m] = valid ? eidx[E + ec] : 0;
    } else {
        int s = eidx[ec];
        for (int j = 0; j < 16; ++j)
            sXS[wave][m][j] = valid ? x[(size_t)s * 16 + j] : 0.0f;
    }
    __syncthreads();

    v8f acc = {};
    const float* wXS = &sXS[wave][m][0];
    const float* wEA = &sEA[wave][m][0];
#pragma unroll
    for (int kk = 0; kk < 68; ++kk) {
        const int c  = (4 * kk) & 15;
        const int i0 = c + 2 * hi;                 // this lane's K index (even)
        float x0 = wXS[i0], x1 = wXS[i0 + 1];
        v2f a, b;
        if (kk < 64) {                              // ea_f * x_i part
            float eaf = wEA[kk >> 2];
            a.x = eaf * x0; a.y = eaf * x1;
        } else {                                    // bias rows: u = x_i
            a.x = x0; a.y = x1;
        }
        const int p = 2 * kk + hi;                  // K-pair index
        b.x = sWp[p * 32 + m * 2 + 0];
        b.y = sWp[p * 32 + m * 2 + 1];
        acc = __builtin_amdgcn_wmma_f32_16x16x4_f32(
            false, a, false, b, (short)0, acc, false, false);
    }

    // Scatter: VGPR v holds row (v + 8*hi), col m of the 16x16 D tile.
#pragma unroll
    for (int v = 0; v < 8; ++v) {
        int mr = v + 8 * hi;
        int er = base + mr;
        if (er < E) {
            int d = sDST[wave][mr];
            atomicAdd(&agg[(size_t)d * 16 + m], acc[v]);
        }
    }
}

// ---------- K2: h = relu(agg + x @ W_root + b_ecc) --------------------------
__global__ void k_hnode(const float* __restrict__ x,
                        const float* __restrict__ W_root,
                        const float* __restrict__ b_ecc,
                        const float* __restrict__ agg,
                        float* __restrict__ h, int N) {
    int i = blockIdx.x * blockDim.x + threadIdx.x;
    if (i >= N * 16) return;
    int n = i >> 4, o = i & 15;
    float s = agg[i] + b_ecc[o];
    const float* xr = x + (size_t)n * 16;
#pragma unroll
    for (int j = 0; j < 16; ++j) s += xr[j] * W_root[j * 16 + o];
    h[i] = fmaxf(s, 0.0f);
}

// ---------- K3: z = h @ W_gat  ([N,16]@[16,128]) via WMMA f32 16x16x4 -------
__global__ void k_zgemm(const float* __restrict__ h,
                        const float* __restrict__ W_gat,
                        float* __restrict__ z, int N) {
    __shared__ float sWg[8 * 256];   // pair-interleaved W_gat: [p=k>>1][col][k&1]
    const int tid = threadIdx.x;
    for (int t = tid; t < 2048; t += 256) {
        int k = t >> 7, col = t & 127;
        sWg[(k >> 1) * 256 + col * 2 + (k & 1)] = W_gat[k * 128 + col];
    }
    __syncthreads();

    const int wave = tid >> 5, lane = tid & 31;
    const int m = lane & 15, hi = lane >> 4;
    const int rowTile = blockIdx.x;      // 16 nodes per block
    const int colTile = wave;            // 8 waves cover 128 outputs
    const int row = rowTile * 16 + m;
    const int rc  = row < N ? row : (N - 1);
    const float* hr = h + (size_t)rc * 16;
    const int col = colTile * 16 + m;

    v8f acc = {};
#pragma unroll
    for (int kk = 0; kk < 4; ++kk) {
        int i0 = 4 * kk + 2 * hi;
        v2f a, b;
        a.x = hr[i0]; a.y = hr[i0 + 1];
        int p = 2 * kk + hi;
        b.x = sWg[p * 256 + col * 2 + 0];
        b.y = sWg[p * 256 + col * 2 + 1];
        acc = __builtin_amdgcn_wmma_f32_16x16x4_f32(
            false, a, false, b, (short)0, acc, false, false);
    }
#pragma unroll
    for (int v = 0; v < 8; ++v) {
        int r = rowTile * 16 + v + 8 * hi;
        if (r < N) z[(size_t)r * 128 + col] = acc[v];
    }
}

// ---------- K4: per-node per-head attention scores --------------------------
__global__ void k_escore(const float* __restrict__ z,
                         const float* __restrict__ a_src,
                         const float* __restrict__ a_dst,
                         float* __restrict__ esrc,
                         float* __restrict__ edst, int N) {
    int i = blockIdx.x * blockDim.x + threadIdx.x;
    if (i >= N * 4) return;
    int n = i >> 2, hh = i & 3;
    const float* zr = z + (size_t)n * 128 + hh * 32;
    const float* as = a_src + hh * 32;
    const float* ad = a_dst + hh * 32;
    float s = 0.f, d = 0.f;
#pragma unroll 8
    for (int j = 0; j < 32; ++j) { float zv = zr[j]; s += zv * as[j]; d += zv * ad[j]; }
    esrc[i] = s; edst[i] = d;
}

// ---------- K5: segment max of leaky-relu logits (atomic on ordered key) ----
__global__ void k_attmax(const int* __restrict__ eidx,
                         const float* __restrict__ esrc,
                         const float* __restrict__ edst,
                         unsigned* __restrict__ maxkey, int E) {
    int i = blockIdx.x * blockDim.x + threadIdx.x;
    if (i >= E * 4) return;
    int e = i >> 2, hh = i & 3;
    int s = eidx[e], d = eidx[E + e];
    float lg = esrc[s * 4 + hh] + edst[d * 4 + hh];
    lg = lg > 0.f ? lg : NEG_SLOPE * lg;
    atomicMax(&maxkey[d * 4 + hh], fkey_enc(lg));
}

// ---------- K6: exp-weights: denom += ex, gatout += ex * z[src] -------------
__global__ void k_attagg(const int* __restrict__ eidx,
                         const float* __restrict__ esrc,
                         const float* __restrict__ edst,
                         const unsigned* __restrict__ maxkey,
                         const float* __restrict__ z,
                         float* __restrict__ denom,
                         float* __restrict__ gatout, int E) {
    int i = blockIdx.x * blockDim.x + threadIdx.x;
    if (i >= E * 4) return;
    int e = i >> 2, hh = i & 3;
    int s = eidx[e], d = eidx[E + e];
    float lg = esrc[s * 4 + hh] + edst[d * 4 + hh];
    lg = lg > 0.f ? lg : NEG_SLOPE * lg;
    float mx = fkey_dec(maxkey[d * 4 + hh]);
    float ex = __expf(lg - mx);
    atomicAdd(&denom[d * 4 + hh], ex);
    const float* zr = z + (size_t)s * 128 + hh * 32;
    float* op = gatout + (size_t)d * 128 + hh * 32;
#pragma unroll 4
    for (int j = 0; j < 32; ++j) atomicAdd(op + j, ex * zr[j]);
}

// ---------- K7: g = relu(gatout/denom + b_gat); y = g @ W_fc + b_fc ---------
__global__ void k_final(const float* __restrict__ gatout,
                        const float* __restrict__ denom,
                        const float* __restrict__ b_gat,
                        const float* __restrict__ W_fc,
                        const float* __restrict__ b_fc,
                        float* __restrict__ out, int N) {
    int wave = threadIdx.x >> 5, lane = threadIdx.x & 31;
    int n = blockIdx.x * 8 + wave;
    if (n >= N) return;
    float acc = 0.f;
#pragma unroll
    for (int t = 0; t < 4; ++t) {
        int j = t * 32 + lane;
        float dn = denom[n * 4 + t] + 1e-16f;
        float g = gatout[(size_t)n * 128 + j] / dn + b_gat[j];
        g = fmaxf(g, 0.0f);
        acc += g * W_fc[j];
    }
    for (int off = 16; off > 0; off >>= 1) acc += __shfl_xor(acc, off, 32);
    if (lane == 0) out[n] = acc + b_fc[0];
}

// ---------------------------------------------------------------------------
extern "C" void kernel_launch(void* const* d_in, const int* in_sizes, int n_in,
                              void* d_out, int out_size, void* d_ws, size_t ws_size,
                              hipStream_t stream) {
    const float* x      = (const float*)d_in[0];
    const int*   eidx   = (const int*)  d_in[1];
    const float* eattr  = (const float*)d_in[2];
    const float* W_nn   = (const float*)d_in[3];
    const float* b_nn   = (const float*)d_in[4];
    const float* W_root = (const float*)d_in[5];
    const float* b_ecc  = (const float*)d_in[6];
    const float* W_gat  = (const float*)d_in[7];
    const float* a_src  = (const float*)d_in[8];
    const float* a_dst  = (const float*)d_in[9];
    const float* b_gat  = (const float*)d_in[10];
    const float* W_fc   = (const float*)d_in[11];
    const float* b_fc   = (const float*)d_in[12];
    float* out = (float*)d_out;

    const int N = in_sizes[0] / 16;
    const int E = in_sizes[1] / 2;

    // Workspace layout (floats): regions needing zeroing come first.
    float* ws = (float*)d_ws;
    size_t off = 0;
    float*    agg    = ws + off;               off += (size_t)N * 16;
    unsigned* maxkey = (unsigned*)(ws + off);  off += (size_t)N * 4;
    float*    denom  = ws + off;               off += (size_t)N * 4;
    float*    gatout = ws + off;               off += (size_t)N * 128;
    const long long zeroCnt = (long long)off;
    float*    hbuf   = ws + off;               off += (size_t)N * 16;
    float*    zbuf   = ws + off;               off += (size_t)N * 128;
    float*    esrc   = ws + off;               off += (size_t)N * 4;
    float*    edst   = ws + off;               off += (size_t)N * 4;

    k_zero<<<2048, 256, 0, stream>>>(ws, zeroCnt);

    int tilesE = (E + 15) / 16;
    k_nnconv<<<(tilesE + 7) / 8, 256, 0, stream>>>(x, eidx, eattr, W_nn, b_nn, agg, N, E);
    k_hnode <<<(N * 16 + 255) / 256, 256, 0, stream>>>(x, W_root, b_ecc, agg, hbuf, N);
    k_zgemm <<<(N + 15) / 16, 256, 0, stream>>>(hbuf, W_gat, zbuf, N);
    k_escore<<<(N * 4 + 255) / 256, 256, 0, stream>>>(zbuf, a_src, a_dst, esrc, edst, N);
    k_attmax<<<(E * 4 + 255) / 256, 256, 0, stream>>>(eidx, esrc, edst, maxkey, E);
    k_attagg<<<(E * 4 + 255) / 256, 256, 0, stream>>>(eidx, esrc, edst, maxkey, zbuf,
                                                      denom, gatout, E);
    k_final <<<(N + 7) / 8, 256, 0, stream>>>(gatout, denom, b_gat, W_fc, b_fc, out, N);
}